// CoTeachingLoss_76974403878992
// MI455X (gfx1250) — compile-verified
//
#include <hip/hip_runtime.h>
#include <hip/hip_bf16.h>

// ---------------- problem constants ----------------
constexpr int N_ROWS = 65536;     // samples per net
constexpr int C      = 1000;      // classes
constexpr int C4     = 250;       // float4 per row (4000 B, 16B aligned stride)
constexpr int TPB    = 128;       // 4 waves (wave32)
constexpr int ROWS_PER_BLK = 16;
constexpr int TOTAL_ROWS   = 2 * N_ROWS;
constexpr int NBLK         = TOTAL_ROWS / ROWS_PER_BLK;   // 8192

// =====================================================================
// Kernel A: per-row cross entropy for both matrices.
// Rows are DMA-staged global->LDS with async b128 loads (ASYNCcnt),
// double buffered so row i+1 streams in while row i is reduced.
// =====================================================================
__global__ __launch_bounds__(TPB)
void ce_rows(const float* __restrict__ y1, const float* __restrict__ y2,
             const int* __restrict__ tgt, float* __restrict__ lossOut)
{
    __shared__ float4 buf[2][2 * TPB];   // 2 x 4096 B row buffers
    __shared__ float  xw[TPB / 32];      // cross-wave reduction scratch

    const int tid  = threadIdx.x;
    const int lane = tid & 31;
    const int wid  = tid >> 5;
    const int row0 = blockIdx.x * ROWS_PER_BLK;

    auto rowPtr = [&](int r) -> const float* {
        return (r < N_ROWS) ? (y1 + (size_t)r * C)
                            : (y2 + (size_t)(r - N_ROWS) * C);
    };

    auto prefetch = [&](int r, int bi) {
        const char* base = (const char*)rowPtr(r);
#pragma unroll
        for (int it = 0; it < 2; ++it) {
            int slot  = it * TPB + tid;          // 0..255
            int gslot = (slot < C4) ? slot : 0;  // clamp tail (no OOB global read)
            unsigned lds = (unsigned)(uintptr_t)(&buf[bi][slot]);
            const char* g = base + (size_t)gslot * 16;
            asm volatile("global_load_async_to_lds_b128 %0, %1, off"
                         :: "v"(lds), "v"(g) : "memory");
        }
    };

    const float L2E = 1.44269504088896340736f;
    const float LN2 = 0.69314718055994530942f;

    prefetch(row0, 0);
    for (int i = 0; i < ROWS_PER_BLK; ++i) {
        const int r = row0 + i;
        if (i + 1 < ROWS_PER_BLK) {
            prefetch(r + 1, (i + 1) & 1);                 // overlap next row DMA
            asm volatile("s_wait_asynccnt 0x2" ::: "memory"); // current row done
        } else {
            asm volatile("s_wait_asynccnt 0x0" ::: "memory");
        }
        __syncthreads();   // all waves' DMA results visible

        const float4* b = buf[i & 1];
        float4 a0 = b[tid];                // slots 0..127, always valid
        float4 a1 = b[tid + TPB];          // slots 128..255; valid if < 250
        const bool v1 = (tid + TPB) < C4;  // tid < 122

        // ---- row max (wave32 shuffle + 4-entry LDS) ----
        float m = fmaxf(fmaxf(a0.x, a0.y), fmaxf(a0.z, a0.w));
        if (v1) m = fmaxf(m, fmaxf(fmaxf(a1.x, a1.y), fmaxf(a1.z, a1.w)));
#pragma unroll
        for (int off = 16; off > 0; off >>= 1)
            m = fmaxf(m, __shfl_xor(m, off, 32));
        if (lane == 0) xw[wid] = m;
        __syncthreads();
        m = fmaxf(fmaxf(xw[0], xw[1]), fmaxf(xw[2], xw[3]));
        __syncthreads();   // xw reuse

        // ---- sum of exp ----
        float s = exp2f((a0.x - m) * L2E) + exp2f((a0.y - m) * L2E)
                + exp2f((a0.z - m) * L2E) + exp2f((a0.w - m) * L2E);
        if (v1)
            s += exp2f((a1.x - m) * L2E) + exp2f((a1.y - m) * L2E)
               + exp2f((a1.z - m) * L2E) + exp2f((a1.w - m) * L2E);
#pragma unroll
        for (int off = 16; off > 0; off >>= 1)
            s += __shfl_xor(s, off, 32);
        if (lane == 0) xw[wid] = s;
        __syncthreads();
        const float S = xw[0] + xw[1] + xw[2] + xw[3];

        if (tid == 0) {
            const int trow = (r < N_ROWS) ? r : (r - N_ROWS);
            const int tg   = tgt[trow];
            const float xt = ((const float*)b)[tg];
            lossOut[r] = log2f(S) * LN2 + m - xt;
        }
        __syncthreads();   // all LDS reads done before next prefetch overwrites
    }
}

// =====================================================================
// Kernel B: means + exact small-loss selection (stable argsort prefix)
// via 4-round byte radix-select on order-preserving float keys, with
// index tie-break, then deterministic masked peer sums.
// =====================================================================
__global__ __launch_bounds__(1024)
void finalize(const float* __restrict__ loss, const int* __restrict__ nkp,
              float* __restrict__ out)
{
    constexpr int N = N_ROWS;
    constexpr int T = 1024;
    __shared__ float    red[T];
    __shared__ unsigned hist[256];
    __shared__ unsigned sh_prefix;
    __shared__ int      sh_rank;
    __shared__ unsigned sh_T[2];
    __shared__ int      sh_need[2];

    const int tid = threadIdx.x;
    const int k   = *nkp;

    auto fkey = [](float f) -> unsigned {
        unsigned u = __float_as_uint(f);
        return u ^ (unsigned)(((int)u >> 31) | (int)0x80000000);
    };
    auto bsum = [&](float v) -> float {
        red[tid] = v; __syncthreads();
        for (int off = T / 2; off > 0; off >>= 1) {
            if (tid < off) red[tid] += red[tid + off];
            __syncthreads();
        }
        float r = red[0]; __syncthreads();
        return r;
    };

    // ---- full means ----
    float p1 = 0.f, p2 = 0.f;
    for (int i = tid; i < N; i += T) { p1 += loss[i]; p2 += loss[N + i]; }
    const float s1 = bsum(p1);
    const float s2 = bsum(p2);

    // ---- radix select: k-th smallest key per loss vector ----
    for (int sel = 0; sel < 2; ++sel) {
        const float* A = loss + sel * N;
        if (tid == 0) { sh_prefix = 0; sh_rank = k; }
        __syncthreads();
        for (int round = 0; round < 4; ++round) {
            const int shift = 24 - 8 * round;
            const unsigned mhi = (round == 0) ? 0u : (0xFFFFFFFFu << (32 - 8 * round));
            if (tid < 256) hist[tid] = 0;
            __syncthreads();
            const unsigned pfx = sh_prefix;
            for (int i = tid; i < N; i += T) {
                unsigned key = fkey(A[i]);
                if ((key & mhi) == pfx) atomicAdd(&hist[(key >> shift) & 255], 1u);
            }
            __syncthreads();
            if (tid == 0) {
                unsigned cum = 0; int rank = sh_rank;
                for (int b = 0; b < 256; ++b) {
                    unsigned c = hist[b];
                    if (cum + c >= (unsigned)rank) {
                        sh_prefix = pfx | ((unsigned)b << shift);
                        sh_rank   = rank - (int)cum;
                        break;
                    }
                    cum += c;
                }
            }
            __syncthreads();
        }
        if (tid == 0) { sh_T[sel] = sh_prefix; sh_need[sel] = sh_rank; }
        __syncthreads();
    }

    // ---- cross peer sums over selected sets ----
    float upd[2];
    for (int sel = 0; sel < 2; ++sel) {
        const float* A = loss + sel * N;         // selecting loss
        const float* P = loss + (1 - sel) * N;   // peer loss to average
        const unsigned Tk = sh_T[sel];
        const int need    = sh_need[sel];

        int ce = 0;
        for (int i = tid; i < N; i += T) if (fkey(A[i]) == Tk) ++ce;
        const int cnt_eq = (int)bsum((float)ce);

        unsigned ithr = 0xFFFFFFFFu;
        const bool use_all = (need >= cnt_eq);   // no index tie-break needed
        if (!use_all) {
            // stable argsort takes the `need` smallest indices among ties
            if (tid == 0) { sh_prefix = 0; sh_rank = need; }
            __syncthreads();
            for (int round = 0; round < 4; ++round) {
                const int shift = 24 - 8 * round;
                const unsigned mhi = (round == 0) ? 0u : (0xFFFFFFFFu << (32 - 8 * round));
                if (tid < 256) hist[tid] = 0;
                __syncthreads();
                const unsigned pfx = sh_prefix;
                for (int i = tid; i < N; i += T) {
                    if (fkey(A[i]) == Tk) {
                        unsigned ui = (unsigned)i;
                        if ((ui & mhi) == pfx) atomicAdd(&hist[(ui >> shift) & 255], 1u);
                    }
                }
                __syncthreads();
                if (tid == 0) {
                    unsigned cum = 0; int rank = sh_rank;
                    for (int b = 0; b < 256; ++b) {
                        unsigned c = hist[b];
                        if (cum + c >= (unsigned)rank) {
                            sh_prefix = pfx | ((unsigned)b << shift);
                            sh_rank   = rank - (int)cum;
                            break;
                        }
                        cum += c;
                    }
                }
                __syncthreads();
            }
            ithr = sh_prefix;
            __syncthreads();
        }

        float ps = 0.f;
        for (int i = tid; i < N; i += T) {
            const unsigned key = fkey(A[i]);
            const bool in_set = (key < Tk) ||
                                (key == Tk && (use_all || (unsigned)i <= ithr));
            if (in_set) ps += P[i];
        }
        upd[sel] = bsum(ps);
    }

    if (tid == 0) {
        out[0] = upd[1] / (float)k;   // loss_1_update: selected by net 2
        out[1] = upd[0] / (float)k;   // loss_2_update: selected by net 1
        out[2] = s1 / (float)N;       // loss_1 mean
        out[3] = s2 / (float)N;       // loss_2 mean
    }
}

// =====================================================================
extern "C" void kernel_launch(void* const* d_in, const int* in_sizes, int n_in,
                              void* d_out, int out_size, void* d_ws, size_t ws_size,
                              hipStream_t stream)
{
    const float* y1  = (const float*)d_in[0];
    const float* y2  = (const float*)d_in[1];
    const int*   tgt = (const int*)d_in[2];
    const int*   nk  = (const int*)d_in[3];
    float*       out = (float*)d_out;
    float*       lossbuf = (float*)d_ws;    // 131072 floats = 512 KB

    ce_rows<<<NBLK, TPB, 0, stream>>>(y1, y2, tgt, lossbuf);
    finalize<<<1, 1024, 0, stream>>>(lossbuf, nk, out);
}